// SeqRecModel_79508434584150
// MI455X (gfx1250) — compile-verified
//
#include <hip/hip_runtime.h>
#include <hip/hip_bf16.h>

// Problem constants (from reference: B=8, S=32, I=64, d=256)
constexpr int kB = 8;
constexpr int kL = 32 * 64;   // 2048 tokens per batch row (S*I)
constexpr int kD = 256;       // feature dim (fp32 -> 1 KB per row)

// ---------------------------------------------------------------------------
// CDNA5 async global<->LDS DMA (gfx1250). Guarded so host pass / toolchains
// without the builtins still compile (register-copy fallback).
// Builtin signature (probe-confirmed): (addrspace(1) int4*, addrspace(3) int4*,
// imm offset, imm cpol).
// ---------------------------------------------------------------------------
#if defined(__has_builtin)
#if __has_builtin(__builtin_amdgcn_global_load_async_to_lds_b128) && \
    __has_builtin(__builtin_amdgcn_global_store_async_from_lds_b128)
#define USE_ASYNC_LDS 1
#endif
#endif

#ifdef USE_ASYNC_LDS
typedef int v4i __attribute__((vector_size(16)));
typedef __attribute__((address_space(1))) v4i* gptr_t;
typedef __attribute__((address_space(3))) v4i* lptr_t;
#if __has_builtin(__builtin_amdgcn_s_wait_asynccnt)
#define WAIT_ASYNC0() __builtin_amdgcn_s_wait_asynccnt(0)
#else
#define WAIT_ASYNC0() asm volatile("s_wait_asynccnt 0" ::: "memory")
#endif
#endif

// ---------------------------------------------------------------------------
// Phase B (listed first so the disasm snippet shows it): one wave32 per token
// row (b, l).
//   - if l >= count[b]: zero output row l   (tail padding)
//   - if mask valid:    copy 1KB source row (b,l) -> output row (b,pos[l])
//     via async global->LDS->global DMA (2x b128 per direction per wave).
// Copy destinations (p < cnt) and zero destinations (l >= cnt) are disjoint,
// so every output row is written exactly once per launch.
// ---------------------------------------------------------------------------
__global__ void scatter_kernel(const float* __restrict__ src,
                               const unsigned char* __restrict__ mask,
                               const int* __restrict__ pos,
                               const int* __restrict__ count,
                               float* __restrict__ out) {
    constexpr int kWaves = 8;               // 256 threads = 8 wave32
#ifdef USE_ASYNC_LDS
    __shared__ float tile[kWaves][kD];      // 1 KB staging row per wave
#endif
    const int lane = threadIdx.x & 31;
    const int wav  = threadIdx.x >> 5;
    const int row  = blockIdx.x * kWaves + wav;   // 0 .. B*L-1
    const int b    = row >> 11;                   // / 2048
    const int l    = row & (kL - 1);

    const int  cnt   = count[b];
    const bool valid = mask[(size_t)b * kL + l] != 0;

    // Tail padding: output row l gets zeros.
    if (l >= cnt) {
        float4 z = {0.f, 0.f, 0.f, 0.f};
        float4* o = (float4*)(out + ((size_t)b * kL + l) * kD) + lane;
        o[0]  = z;       // bytes [lane*16, +16)
        o[32] = z;       // bytes [512 + lane*16, +16)
    }

    // Stable left-pack: copy source row l to output slot pos[l].
    if (valid) {
        const int p = pos[(size_t)b * kL + l];
        const float* g = src + ((size_t)b * kL + l) * kD + lane * 4;  // 16B/lane
        float*       o = out + ((size_t)b * kL + p) * kD + lane * 4;
#ifdef USE_ASYNC_LDS
        // Wave-uniform branch -> EXEC is all-ones here.
        gptr_t gsp = (gptr_t)g;                 // C-style cast drops const + addrspacecast
        gptr_t gdp = (gptr_t)o;
        lptr_t lp  = (lptr_t)&tile[wav][lane * 4];
        // 2 x 512B: the instruction offset applies to both global and LDS sides.
        __builtin_amdgcn_global_load_async_to_lds_b128(gsp, lp, 0, 0);
        __builtin_amdgcn_global_load_async_to_lds_b128(gsp, lp, 512, 0);
        WAIT_ASYNC0();   // loads must land in LDS before the stores read it
        __builtin_amdgcn_global_store_async_from_lds_b128(gdp, lp, 0, 0);
        __builtin_amdgcn_global_store_async_from_lds_b128(gdp, lp, 512, 0);
        // s_endpgm performs an implicit wait-idle; stores drain before exit.
#else
        const float4* gs = (const float4*)g;
        float4*       go = (float4*)o;
        float4 a0 = gs[0];
        float4 a1 = gs[32];
        go[0]  = a0;
        go[32] = a1;
#endif
    }
}

// ---------------------------------------------------------------------------
// Phase A: per-batch stable exclusive scan of the valid mask.
//   pos[b*L + l]  = # of valid tokens before l   (destination slot if valid)
//   count[b]      = total # valid tokens in batch row b
// One block of 256 threads per batch row; 8 mask bytes per thread.
// (jnp bool arrays are 1 byte per element -> unsigned char.)
// ---------------------------------------------------------------------------
__global__ void scan_kernel(const unsigned char* __restrict__ mask,
                            int* __restrict__ pos,
                            int* __restrict__ count) {
    __shared__ int sums[256];
    const int b = blockIdx.x;
    const int t = threadIdx.x;

    const unsigned char* m = mask + (size_t)b * kL + t * 8;
    int v[8];
    int s = 0;
#pragma unroll
    for (int i = 0; i < 8; ++i) {
        v[i] = m[i] ? 1 : 0;
        s += v[i];
    }
    sums[t] = s;
    __syncthreads();

    // Inclusive Hillis-Steele scan over the 256 per-thread partials.
    for (int off = 1; off < 256; off <<= 1) {
        int x = sums[t];
        int y = (t >= off) ? sums[t - off] : 0;
        __syncthreads();
        sums[t] = x + y;
        __syncthreads();
    }

    if (t == 255) count[b] = sums[255];

    int excl = (t == 0) ? 0 : sums[t - 1];
    int* prow = pos + (size_t)b * kL + t * 8;
#pragma unroll
    for (int i = 0; i < 8; ++i) {
        prow[i] = excl;
        excl += v[i];
    }
}

extern "C" void kernel_launch(void* const* d_in, const int* in_sizes, int n_in,
                              void* d_out, int out_size, void* d_ws, size_t ws_size,
                              hipStream_t stream) {
    const float*         ffn  = (const float*)d_in[0];          // [8,32,64,256] fp32
    const unsigned char* mask = (const unsigned char*)d_in[1];  // [8,32,64] bool (1B)
    float*               out  = (float*)d_out;                  // [8,2048,256] fp32

    // Workspace: pos[B*L] ints + count[B] ints = 64 KB + 32 B.
    int* pos = (int*)d_ws;
    int* cnt = pos + (size_t)kB * kL;

    // Phase A: per-batch exclusive scan of the mask.
    scan_kernel<<<dim3(kB), dim3(256), 0, stream>>>(mask, pos, cnt);

    // Phase B: one wave per token row; 8 waves per block.
    scatter_kernel<<<dim3(kB * kL / 8), dim3(256), 0, stream>>>(ffn, mask, pos, cnt, out);
}